// CombinedVirtualNodeEnvEncoder_2602750181779
// MI455X (gfx1250) — compile-verified
//
#include <hip/hip_runtime.h>
#include <math.h>

typedef __attribute__((ext_vector_type(2))) float v2f;
typedef __attribute__((ext_vector_type(8))) float v8f;

#define NODES 100000
#define EDGES 1600000
#define CH    128   // in/out channels

// ---------------------------------------------------------------------------
// Kernel 1: in-degree per node: deg[col[e]] += 1
// ---------------------------------------------------------------------------
__global__ void degree_kernel(const int* __restrict__ eidx, float* __restrict__ deg) {
    for (int e = blockIdx.x * blockDim.x + threadIdx.x; e < EDGES;
         e += gridDim.x * blockDim.x) {
        int c = eidx[EDGES + e];
        atomicAdd(&deg[c], 1.0f);
    }
}

// ---------------------------------------------------------------------------
// Kernel 2: hi[col] += val * x[row]; one wave32 per edge, 4 floats per lane
// val = 1/sqrt(deg[col]*deg[row]) with non-finite -> 0
// ---------------------------------------------------------------------------
__global__ __launch_bounds__(256) void scatter_kernel(
    const float* __restrict__ x, const int* __restrict__ eidx,
    const float* __restrict__ deg, float* __restrict__ hi) {
    const int lane = threadIdx.x & 31;
    const int wave = threadIdx.x >> 5;
    const long long e = (long long)blockIdx.x * 8 + wave;
    if (e >= EDGES) return;
    const int r = eidx[e];
    const int c = eidx[EDGES + e];
    const float p = deg[r] * deg[c];
    const float val = (p > 0.0f) ? rsqrtf(p) : 0.0f;   // inf/nan -> 0 path
    const float4 xv = ((const float4*)(x + (size_t)r * CH))[lane];
    float* hp = hi + (size_t)c * CH + lane * 4;
    atomicAdd(hp + 0, xv.x * val);
    atomicAdd(hp + 1, xv.y * val);
    atomicAdd(hp + 2, xv.z * val);
    atomicAdd(hp + 3, xv.w * val);
}

// ---------------------------------------------------------------------------
// Kernel 3: column sums of x -> gp_sum[128]
// ---------------------------------------------------------------------------
__global__ __launch_bounds__(128) void col_sum_kernel(
    const float* __restrict__ x, float* __restrict__ gp_sum) {
    const int t = threadIdx.x;
    float s = 0.0f;
    for (int r = blockIdx.x; r < NODES; r += gridDim.x)
        s += x[(size_t)r * CH + t];
    atomicAdd(&gp_sum[t], s);
}

// ---------------------------------------------------------------------------
// Kernel 4: tiny MLP on mean vector + fold (1-w)*ge@fc_w + fc_b into g_vec,
// w = sigmoid(alpha) into w_scalar. Single block, 128 threads.
// ---------------------------------------------------------------------------
__global__ __launch_bounds__(128) void prep_kernel(
    const float* __restrict__ gp_sum,
    const float* __restrict__ w1, const float* __restrict__ b1,
    const float* __restrict__ w2, const float* __restrict__ b2,
    const float* __restrict__ alpha,
    const float* __restrict__ fc_w, const float* __restrict__ fc_b,
    float* __restrict__ g_vec, float* __restrict__ w_scalar) {
    __shared__ float sg[CH], sh[CH], se[CH];
    const int t = threadIdx.x;
    sg[t] = gp_sum[t] * (1.0f / (float)NODES);
    __syncthreads();
    float a = b1[t];
    for (int i = 0; i < CH; ++i) a += sg[i] * w1[i * CH + t];
    sh[t] = fmaxf(a, 0.0f);
    __syncthreads();
    a = b2[t];
    for (int i = 0; i < CH; ++i) a += sh[i] * w2[i * CH + t];
    se[t] = a;
    __syncthreads();
    const float wv = 1.0f / (1.0f + expf(-alpha[0]));
    a = 0.0f;
    for (int i = 0; i < CH; ++i) a += se[i] * fc_w[i * CH + t];
    g_vec[t] = (1.0f - wv) * a + fc_b[t];
    if (t == 0) w_scalar[0] = wv;
}

// ---------------------------------------------------------------------------
// Kernel 5: WMMA fp32 GEMM:  out[nRows x 128] = scale * (A[nRows x 128] @ B) + gvec
// B (128x128) staged in LDS. 8 waves/block, each wave: 16 rows x 128 cols,
// K=128 via 32x V_WMMA_F32_16X16X4_F32. A fragment reused across 8 B tiles.
// ---------------------------------------------------------------------------
__global__ __launch_bounds__(256) void wmma_gemm_kernel(
    const float* __restrict__ A, const float* __restrict__ B,
    float* __restrict__ out, int nRows,
    const float* __restrict__ scale_ptr, const float* __restrict__ gvec) {
    __shared__ float Bs[CH * CH];                       // 64 KB
    {   // coalesced B load, float4
        float4* dst = (float4*)Bs;
        const float4* src = (const float4*)B;
        for (int i = threadIdx.x; i < CH * CH / 4; i += 256) dst[i] = src[i];
    }
    __syncthreads();

    const int lane  = threadIdx.x & 31;
    const int wave  = threadIdx.x >> 5;
    const int lm    = lane & 15;            // M (for A) / N (for B,C) within tile
    const int khalf = (lane >> 4) << 1;     // 0 or 2: K half-select per ISA layout
    const int rowBase = blockIdx.x * 128 + wave * 16;
    const int arow  = rowBase + lm;
    const float* ap = A + (size_t)(arow < nRows ? arow : 0) * CH; // clamp loads

    v8f acc[8] = {};                        // 8 tiles of 16x16 f32 accum

    for (int k0 = 0; k0 < CH; k0 += 4) {
        // A fragment 16x4: lanes0-15 -> K=k0,k0+1 ; lanes16-31 -> K=k0+2,k0+3
        v2f a;
        a.x = ap[k0 + khalf];
        a.y = ap[k0 + khalf + 1];
        const float* b0 = Bs + (k0 + khalf) * CH;       // row K
        const float* b1r = b0 + CH;                     // row K+1
#pragma unroll
        for (int j = 0; j < 8; ++j) {
            // B fragment 4x16: row striped across lanes within VGPR
            v2f b;
            b.x = b0[j * 16 + lm];
            b.y = b1r[j * 16 + lm];
            acc[j] = __builtin_amdgcn_wmma_f32_16x16x4_f32(
                false, a, false, b, (short)0, acc[j], false, false);
        }
    }

    const float scale = scale_ptr ? *scale_ptr : 1.0f;
    const int rofs = (lane >> 4) << 3;      // lanes16-31 hold M = v+8
#pragma unroll
    for (int j = 0; j < 8; ++j) {
        const int col = j * 16 + lm;
        const float gadd = gvec ? gvec[col] : 0.0f;
#pragma unroll
        for (int v = 0; v < 8; ++v) {
            const int r = rowBase + rofs + v;
            if (r < nRows)
                out[(size_t)r * CH + col] = scale * acc[j][v] + gadd;
        }
    }
}

// ---------------------------------------------------------------------------
extern "C" void kernel_launch(void* const* d_in, const int* in_sizes, int n_in,
                              void* d_out, int out_size, void* d_ws, size_t ws_size,
                              hipStream_t stream) {
    const float* x      = (const float*)d_in[0];
    const int*   eidx   = (const int*)  d_in[1];
    const float* wloc   = (const float*)d_in[2];
    const float* w1     = (const float*)d_in[3];
    const float* b1     = (const float*)d_in[4];
    const float* w2     = (const float*)d_in[5];
    const float* b2     = (const float*)d_in[6];
    const float* alpha  = (const float*)d_in[7];
    const float* fc_w   = (const float*)d_in[8];
    const float* fc_b   = (const float*)d_in[9];
    float* out = (float*)d_out;

    // workspace layout (floats)
    float* ws      = (float*)d_ws;
    float* deg     = ws;                                   // NODES (padded 100352)
    float* hi      = ws + 100352;                          // NODES*CH
    float* gp_sum  = hi + (size_t)NODES * CH;              // 128
    float* g_vec   = gp_sum + CH;                          // 128
    float* w_scal  = g_vec + CH;                           // 1 (padded 128)
    float* Mfused  = w_scal + 128;                         // 128*128

    // zero deg + hi + gp_sum (everything up to Mfused)
    size_t zbytes = (size_t)((char*)Mfused - (char*)ws);
    hipMemsetAsync(d_ws, 0, zbytes, stream);

    degree_kernel<<<2048, 256, 0, stream>>>(eidx, deg);
    scatter_kernel<<<EDGES / 8, 256, 0, stream>>>(x, eidx, deg, hi);
    col_sum_kernel<<<256, 128, 0, stream>>>(x, gp_sum);
    prep_kernel<<<1, 128, 0, stream>>>(gp_sum, w1, b1, w2, b2, alpha,
                                       fc_w, fc_b, g_vec, w_scal);
    // Mfused = weight_local @ fc_w   (128x128, one block)
    wmma_gemm_kernel<<<1, 256, 0, stream>>>(wloc, fc_w, Mfused, CH,
                                            nullptr, nullptr);
    // out = sigmoid(alpha) * (hi @ Mfused) + g_vec
    wmma_gemm_kernel<<<(NODES + 127) / 128, 256, 0, stream>>>(
        hi, Mfused, out, NODES, w_scal, g_vec);
}